// GAT_58454504899116
// MI455X (gfx1250) — compile-verified
//
#include <hip/hip_runtime.h>
#include <hip/hip_bf16.h>

typedef __attribute__((ext_vector_type(2))) float v2f;
typedef __attribute__((ext_vector_type(8))) float v8f;

// ---------------------------------------------------------------------------
// Utility: zero fill (grid-stride)
// ---------------------------------------------------------------------------
__global__ void zero_fill_k(float* __restrict__ p, long long n) {
    long long i = (long long)blockIdx.x * blockDim.x + threadIdx.x;
    long long stride = (long long)gridDim.x * blockDim.x;
    for (; i < n; i += stride) p[i] = 0.0f;
}

// ---------------------------------------------------------------------------
// Build extended weights: Wext[128, ncols+2H] = [ W | W@att_src | W@att_dst ]
// so one GEMM produces xw AND both attention dot products.
// ---------------------------------------------------------------------------
__global__ void build_wext_k(const float* __restrict__ W,      // [128, ncols]
                             const float* __restrict__ att_s,  // [H*C]
                             const float* __restrict__ att_d,  // [H*C]
                             float* __restrict__ Wext,         // [128, ncols+2H]
                             int ncols, int H, int C) {
    const int next = ncols + 2 * H;
    int i = blockIdx.x * blockDim.x + threadIdx.x;
    if (i >= 128 * next) return;
    int f = i / next, c = i % next;
    float v;
    if (c < ncols) {
        v = W[f * ncols + c];
    } else if (c < ncols + H) {
        int h = c - ncols; v = 0.0f;
        for (int cc = 0; cc < C; ++cc) v += W[f * ncols + h * C + cc] * att_s[h * C + cc];
    } else {
        int h = c - ncols - H; v = 0.0f;
        for (int cc = 0; cc < C; ++cc) v += W[f * ncols + h * C + cc] * att_d[h * C + cc];
    }
    Wext[i] = v;
}

// ---------------------------------------------------------------------------
// WMMA fp32 GEMM with fused attention-dot columns.
//   A[N,128] @ Wext[128, ncols+2H]  ->  xw[N,ncols], as[N,H], ad[N,H]
// CT col-tiles of 16; block = CT waves; both operands staged in LDS.
// W is staged ONCE per block and reused across RPB row tiles (amortizes the
// ~74KB LDS fill across 128 rows). B is pair-swizzled so each fragment is one
// ds_load_b64: element (k, n) lives at sW[ ((k>>1)*COLS + n)*2 + (k&1) ].
// ---------------------------------------------------------------------------
template <int CT, int RPB>
__global__ void gemm128_wmma_k(const float* __restrict__ A,
                               const float* __restrict__ Wext,
                               float* __restrict__ xw,
                               float* __restrict__ as_,
                               float* __restrict__ ad_,
                               int N, int ncols, int H) {
    constexpr int COLS = CT * 16;
    __shared__ float sA[16 * 128];
    __shared__ float sW[128 * COLS];

    const int tid  = threadIdx.x;
    const int lane = tid & 31;
    const int wave = tid >> 5;
    const int next = ncols + 2 * H;          // real column count of Wext
    const int nRowTiles = (N + 15) >> 4;

    // Stage Wext pair-swizzled (zero-pad cols beyond next). Once per block.
    for (int j = tid; j < 128 * COLS; j += blockDim.x) {
        int klow = j & 1;
        int t    = j >> 1;
        int n    = t % COLS;
        int kp   = t / COLS;
        int k    = kp * 2 + klow;
        sW[j] = (n < next) ? Wext[k * next + n] : 0.0f;
    }

    const int m    = lane & 15;          // A row / B,C col within tile
    const int koff = (lane >> 4) << 1;   // lanes 0-15 -> K{0,1}; 16-31 -> K{2,3}
    const int ncol = wave * 16 + m;      // global output column

    // Route output column to xw / a_src / a_dst (loop-invariant).
    float* dst = nullptr; int ld = 0;
    if (ncol < ncols)               { dst = xw  + ncol;               ld = ncols; }
    else if (ncol < ncols + H)      { dst = as_ + (ncol - ncols);     ld = H; }
    else if (ncol < ncols + 2 * H)  { dst = ad_ + (ncol - ncols - H); ld = H; }
    const int mbase = (lane < 16) ? 0 : 8;

    for (int it = 0; it < RPB; ++it) {
        const int rtile = blockIdx.x * RPB + it;
        if (rtile >= nRowTiles) break;
        const int row0 = rtile * 16;

        // Fence previous iteration's compute, then stage the A tile.
        __syncthreads();
        for (int i = tid; i < 16 * 128; i += blockDim.x) {
            int r = i >> 7, c = i & 127;
            sA[i] = (row0 + r < N) ? A[(size_t)(row0 + r) * 128 + c] : 0.0f;
        }
        __syncthreads();

        v8f acc = {};
#pragma unroll
        for (int kk = 0; kk < 128; kk += 4) {
            const int k0 = kk + koff;    // even
            const float2 aa = *(const float2*)&sA[m * 128 + k0];
            const float2 bb = *(const float2*)&sW[((k0 >> 1) * COLS + ncol) * 2];
            v2f a, b;
            a.x = aa.x; a.y = aa.y;
            b.x = bb.x; b.y = bb.y;
            acc = __builtin_amdgcn_wmma_f32_16x16x4_f32(
                false, a, false, b, (short)0, acc, false, false);
        }

        if (dst) {
#pragma unroll
            for (int r = 0; r < 8; ++r) {
                int row = row0 + mbase + r;
                if (row < N) dst[(size_t)row * ld] = acc[r];
            }
        }
    }
}

// ---------------------------------------------------------------------------
// Edge helpers
// ---------------------------------------------------------------------------
__device__ __forceinline__ void edge_nodes(const long long* __restrict__ ei,
                                           long long E, long long e,
                                           int& s, int& d) {
    if (e < E) { s = (int)ei[e]; d = (int)ei[E + e]; }
    else       { s = d = (int)(e - E); }            // appended self-loop
}

__device__ __forceinline__ float lrelu(float v) { return v > 0.0f ? v : 0.2f * v; }

// Monotone float -> uint map for atomic max over signed floats.
__device__ __forceinline__ unsigned fmap(float f) {
    unsigned u = __float_as_uint(f);
    return (u & 0x80000000u) ? ~u : (u | 0x80000000u);
}
__device__ __forceinline__ float funmap(unsigned u) {
    return __uint_as_float((u & 0x80000000u) ? (u ^ 0x80000000u) : ~u);
}

// Pass 1: segment max (uint-mapped atomicMax per (dst, h)).
template <int H>
__global__ void edge_max_k(const long long* __restrict__ ei, long long E, long long EP,
                           const float* __restrict__ as, const float* __restrict__ ad,
                           unsigned* __restrict__ maxu) {
    long long e = (long long)blockIdx.x * blockDim.x + threadIdx.x;
    if (e >= EP) return;
    int s, d; edge_nodes(ei, E, e, s, d);
#pragma unroll
    for (int h = 0; h < H; ++h) {
        float v = lrelu(as[s * H + h] + ad[d * H + h]);
        atomicMax(&maxu[d * H + h], fmap(v));
    }
}

// Pass 2: e_exp = exp(e - max[dst]); store per-edge, atomic-sum denom per dst.
template <int H>
__global__ void edge_exp_k(const long long* __restrict__ ei, long long E, long long EP,
                           const float* __restrict__ as, const float* __restrict__ ad,
                           const unsigned* __restrict__ maxu,
                           float* __restrict__ denom, float* __restrict__ eexp) {
    long long e = (long long)blockIdx.x * blockDim.x + threadIdx.x;
    if (e >= EP) return;
    int s, d; edge_nodes(ei, E, e, s, d);
#pragma unroll
    for (int h = 0; h < H; ++h) {
        float v  = lrelu(as[s * H + h] + ad[d * H + h]);
        float ex = expf(v - funmap(maxu[d * H + h]));
        eexp[e * H + h] = ex;
        atomicAdd(&denom[d * H + h], ex);
    }
}

// Pass 3 (layer 1): wave per edge; lane handles 4 channels (float4 gather,
// 4 atomic f32 adds). H=4, C=32, F=128; head = lane>>3.
__global__ void edge_agg1_k(const long long* __restrict__ ei, long long E, long long EP,
                            const float* __restrict__ xw,     // [N,128]
                            const float* __restrict__ eexp,   // [EP,4]
                            const float* __restrict__ denom,  // [N,4]
                            float* __restrict__ out) {        // [N,128]
    long long ew = (long long)blockIdx.x * 8 + (threadIdx.x >> 5);
    if (ew >= EP) return;
    const int lane = threadIdx.x & 31;
    int s, d; edge_nodes(ei, E, ew, s, d);
    const int h = lane >> 3;
    const float alpha = eexp[ew * 4 + h] / denom[d * 4 + h];
    const float4 v = ((const float4*)(xw + (size_t)s * 128))[lane];
    float* o = out + (size_t)d * 128 + lane * 4;
    atomicAdd(o + 0, v.x * alpha);
    atomicAdd(o + 1, v.y * alpha);
    atomicAdd(o + 2, v.z * alpha);
    atomicAdd(o + 3, v.w * alpha);
}

// Pass 3 (layer 2): 8 lanes per edge, 5 channels each. H=1, C=40.
__global__ void edge_agg2_k(const long long* __restrict__ ei, long long E, long long EP,
                            const float* __restrict__ xw,     // [N,40]
                            const float* __restrict__ eexp,   // [EP]
                            const float* __restrict__ denom,  // [N]
                            float* __restrict__ out) {        // [N,40]
    long long ew = (long long)blockIdx.x * 32 + (threadIdx.x >> 3);
    if (ew >= EP) return;
    const int g = threadIdx.x & 7;
    int s, d; edge_nodes(ei, E, ew, s, d);
    const float alpha = eexp[ew] / denom[d];
    const float* v = xw + (size_t)s * 40 + g * 5;
    float* o = out + (size_t)d * 40 + g * 5;
#pragma unroll
    for (int j = 0; j < 5; ++j) atomicAdd(o + j, v[j] * alpha);
}

// h = relu(agg + b1)  (in place on the aggregation buffer)
__global__ void bias_relu_k(float* __restrict__ h, const float* __restrict__ b,
                            long long N, int F) {
    long long i = (long long)blockIdx.x * blockDim.x + threadIdx.x;
    if (i >= N * F) return;
    float v = h[i] + b[(int)(i % F)];
    h[i] = v > 0.0f ? v : 0.0f;
}

// out += b2
__global__ void bias_add_k(float* __restrict__ o, const float* __restrict__ b,
                           long long N, int F) {
    long long i = (long long)blockIdx.x * blockDim.x + threadIdx.x;
    if (i >= N * F) return;
    o[i] += b[(int)(i % F)];
}

// ---------------------------------------------------------------------------
// Orchestration
// ---------------------------------------------------------------------------
extern "C" void kernel_launch(void* const* d_in, const int* in_sizes, int n_in,
                              void* d_out, int out_size, void* d_ws, size_t ws_size,
                              hipStream_t stream) {
    (void)n_in; (void)out_size; (void)ws_size;
    const float*     x     = (const float*)d_in[0];
    const long long* ei    = (const long long*)d_in[1];
    const float*     W1    = (const float*)d_in[2];
    const float*     at_s1 = (const float*)d_in[3];
    const float*     at_d1 = (const float*)d_in[4];
    const float*     b1    = (const float*)d_in[5];
    const float*     W2    = (const float*)d_in[6];
    const float*     at_s2 = (const float*)d_in[7];
    const float*     at_d2 = (const float*)d_in[8];
    const float*     b2    = (const float*)d_in[9];
    float* out = (float*)d_out;

    const long long N  = in_sizes[0] / 128;
    const long long E  = in_sizes[1] / 2;
    const long long EP = E + N;            // with self-loops

    // Workspace carve-up.
    float* ws = (float*)d_ws;
    float*    xw1    = ws;              ws += N * 128;
    float*    a_src1 = ws;              ws += N * 4;
    float*    a_dst1 = ws;              ws += N * 4;
    unsigned* maxu1  = (unsigned*)ws;   ws += N * 4;
    float*    denom1 = ws;              ws += N * 4;
    float*    eexp1  = ws;              ws += EP * 4;
    float*    hbuf   = ws;              ws += N * 128;   // agg1 -> relu in place
    float*    xw2    = ws;              ws += N * 40;
    float*    a_src2 = ws;              ws += N;
    float*    a_dst2 = ws;              ws += N;
    unsigned* maxu2  = (unsigned*)ws;   ws += N;
    float*    denom2 = ws;              ws += N;
    float*    eexp2  = ws;              ws += EP;
    float*    Wext1  = ws;              ws += 128 * (128 + 8);  // [W1|W1@as|W1@ad]
    float*    Wext2  = ws;              /* ws += 128 * (40 + 2); */

    const unsigned zgrid = 2048;
    // Zero: [maxu1|denom1] (contiguous), hbuf, [maxu2|denom2] (contiguous), out.
    zero_fill_k<<<zgrid, 256, 0, stream>>>((float*)maxu1, N * 8);
    zero_fill_k<<<zgrid, 256, 0, stream>>>(hbuf, N * 128);
    zero_fill_k<<<zgrid, 256, 0, stream>>>((float*)maxu2, N * 2);
    zero_fill_k<<<zgrid, 256, 0, stream>>>(out, N * 40);

    const long long rowTiles = (N + 15) / 16;
    const unsigned gemmBlocks = (unsigned)((rowTiles + 7) / 8);   // RPB = 8

    // ---- Layer 1 (ncols=128, H=4, C=32; ext=136 -> CT=9) ----
    build_wext_k<<<(128 * 136 + 255) / 256, 256, 0, stream>>>(
        W1, at_s1, at_d1, Wext1, 128, 4, 32);
    gemm128_wmma_k<9, 8><<<gemmBlocks, 9 * 32, 0, stream>>>(
        x, Wext1, xw1, a_src1, a_dst1, (int)N, 128, 4);
    edge_max_k<4><<<(unsigned)((EP + 255) / 256), 256, 0, stream>>>(
        ei, E, EP, a_src1, a_dst1, maxu1);
    edge_exp_k<4><<<(unsigned)((EP + 255) / 256), 256, 0, stream>>>(
        ei, E, EP, a_src1, a_dst1, maxu1, denom1, eexp1);
    edge_agg1_k<<<(unsigned)((EP + 7) / 8), 256, 0, stream>>>(
        ei, E, EP, xw1, eexp1, denom1, hbuf);
    bias_relu_k<<<(unsigned)((N * 128 + 255) / 256), 256, 0, stream>>>(hbuf, b1, N, 128);

    // ---- Layer 2 (ncols=40, H=1, C=40; ext=42 -> CT=3) ----
    build_wext_k<<<(128 * 42 + 255) / 256, 256, 0, stream>>>(
        W2, at_s2, at_d2, Wext2, 40, 1, 40);
    gemm128_wmma_k<3, 8><<<gemmBlocks, 3 * 32, 0, stream>>>(
        hbuf, Wext2, xw2, a_src2, a_dst2, (int)N, 40, 1);
    edge_max_k<1><<<(unsigned)((EP + 255) / 256), 256, 0, stream>>>(
        ei, E, EP, a_src2, a_dst2, maxu2);
    edge_exp_k<1><<<(unsigned)((EP + 255) / 256), 256, 0, stream>>>(
        ei, E, EP, a_src2, a_dst2, maxu2, denom2, eexp2);
    edge_agg2_k<<<(unsigned)((EP + 31) / 32), 256, 0, stream>>>(
        ei, E, EP, xw2, eexp2, denom2, out);
    bias_add_k<<<(unsigned)((N * 40 + 255) / 256), 256, 0, stream>>>(out, b2, N, 40);
}